// RoIAlignMax_58866821759168
// MI455X (gfx1250) — compile-verified
//
#include <hip/hip_runtime.h>

// CDNA5 / gfx1250 ROI-Align + 2x2 s1 max-pool.
// Separable bilinear: V[c] = Wy^T * F[c] * Wx.
//   Stage 1 (VALU, coalesced row loads): T[gy][x][c] = wy0*F[c][iy0][x] + wy1*F[c][iy1][x]
//   Stage 2 (WMMA f32 16x16x4, K=64):    V[c][gy][gx] = sum_x T[gy][x][c] * Wx[x][gx]
//   Stage 3: 2x2 stride-1 max pool from LDS, coalesced global store.

typedef __attribute__((ext_vector_type(2))) float v2f;
typedef __attribute__((ext_vector_type(8))) float v8f;

#define RA_AH 8
#define RA_AW 8
#define RA_SCALE 0.0625f
#define RA_C 256
#define RA_H 64
#define RA_W 64
#define RA_CCHUNK 64            // channels processed per pass
#define RA_CPAD (RA_CCHUNK + 1) // LDS bank-conflict padding

__global__ __launch_bounds__(256)
void roialign_maxpool_wmma(const float* __restrict__ feat,
                           const float* __restrict__ rois,
                           float* __restrict__ out,
                           int nrois)
{
    __shared__ float sT[RA_AH][RA_W][RA_CPAD]; // 8*64*65*4 = 133,120 B
    __shared__ float sWx[RA_W][16];            // 4,096 B (N padded 8->16)
    __shared__ float sV[RA_CCHUNK][RA_AH][RA_AW]; // 16,384 B

    const int n = blockIdx.x;
    if (n >= nrois) return;

    const int t    = threadIdx.x;
    const int lane = t & 31;
    const int wid  = t >> 5;     // 8 waves
    const int mn   = lane & 15;  // M (channels) for A/D, N (gx) for B/D
    const int half = lane >> 4;  // 0: lanes 0-15, 1: lanes 16-31

    // ---- ROI parameters (redundant per thread; scalar-broadcast loads) ----
    const float rb = rois[(size_t)n * 5 + 0];
    const float x1 = rois[(size_t)n * 5 + 1] * RA_SCALE;
    const float y1 = rois[(size_t)n * 5 + 2] * RA_SCALE;
    const float x2 = rois[(size_t)n * 5 + 3] * RA_SCALE;
    const float y2 = rois[(size_t)n * 5 + 4] * RA_SCALE;
    const int   b  = (int)rb;
    const float bin_h = (y2 - y1) / (float)(RA_AH - 1);
    const float bin_w = (x2 - x1) / (float)(RA_AW - 1);

    // ---- Build Wx (64 x 16), two nonzeros per valid gx column ----
    for (int i = t; i < RA_W * 16; i += 256) (&sWx[0][0])[i] = 0.0f;
    __syncthreads();
    if (t < RA_AW) {
        const float xs = x1 + bin_w * (float)t;
        const bool  vx = (xs >= 0.0f) && (xs < (float)RA_W);
        const float x0 = floorf(xs);
        const float lx = xs - x0;
        int ix0 = (int)x0;
        ix0 = ix0 < 0 ? 0 : (ix0 > RA_W - 1 ? RA_W - 1 : ix0);
        const int ix1 = ix0 + 1 > RA_W - 1 ? RA_W - 1 : ix0 + 1;
        if (vx) {
            sWx[ix0][t] += 1.0f - lx;  // same thread, sequential: ix0==ix1 OK
            sWx[ix1][t] += lx;
        }
    }

    const int xcol = t & 63;   // stage-1: x column this thread loads
    const int cgrp = t >> 6;   // stage-1: channel group 0..3

    for (int chunk = 0; chunk < RA_C; chunk += RA_CCHUNK) {
        __syncthreads(); // Wx ready (1st iter); sV consumed (later iters)

        // ---- Stage 1: y-interpolation, fully coalesced 64-wide row loads ----
        for (int gy = 0; gy < RA_AH; ++gy) {
            const float ysv = y1 + bin_h * (float)gy;
            const bool  vy  = (ysv >= 0.0f) && (ysv < (float)RA_H);
            const float y0  = floorf(ysv);
            const float ly  = ysv - y0;
            int iy0 = (int)y0;
            iy0 = iy0 < 0 ? 0 : (iy0 > RA_H - 1 ? RA_H - 1 : iy0);
            const int   iy1 = iy0 + 1 > RA_H - 1 ? RA_H - 1 : iy0 + 1;
            const float wy0 = vy ? (1.0f - ly) : 0.0f;
            const float wy1 = vy ? ly : 0.0f;
            for (int cc = cgrp; cc < RA_CCHUNK; cc += 4) {
                const int c = chunk + cc;
                const float* base = feat + (((size_t)b * RA_C + c) * RA_H) * RA_W;
                const float f0 = base[iy0 * RA_W + xcol];
                const float f1 = base[iy1 * RA_W + xcol];
                sT[gy][xcol][cc] = wy0 * f0 + wy1 * f1;
            }
        }
        __syncthreads();

        // ---- Stage 2: x-interpolation as WMMA f32 16x16x4, K = 64 ----
        // wave w owns grid row gy = w. A: 16 channels x 4 xs. B: 4 xs x 16 gx.
        const int gy = wid;

        // Preload all 16 B fragments (Wx is shared across channel tiles).
        v2f bfrag[16];
        for (int kb = 0; kb < 16; ++kb) {
            const int xb = kb * 4 + 2 * half;     // reg .x -> K=xb, .y -> K=xb+1
            v2f bf;
            bf.x = sWx[xb + 0][mn];
            bf.y = sWx[xb + 1][mn];
            bfrag[kb] = bf;
        }

        for (int ct = 0; ct < RA_CCHUNK / 16; ++ct) {
            const int cbase = ct * 16;
            v8f acc = {};
            for (int kb = 0; kb < 16; ++kb) {
                const int xb = kb * 4 + 2 * half;
                v2f af;
                af.x = sT[gy][xb + 0][cbase + mn];
                af.y = sT[gy][xb + 1][cbase + mn];
                acc = __builtin_amdgcn_wmma_f32_16x16x4_f32(
                    /*neg_a=*/false, af, /*neg_b=*/false, bfrag[kb],
                    /*c_mod=*/(short)0, acc, /*reuse_a=*/false, /*reuse_b=*/false);
            }
            // D layout: reg r at lane L -> (c = cbase + r + 8*half, gx = mn)
            if (mn < RA_AW) {
                #pragma unroll
                for (int r = 0; r < 8; ++r)
                    sV[cbase + r + 8 * half][gy][mn] = acc[r];
            }
        }
        __syncthreads();

        // ---- Stage 3: 2x2 stride-1 max pool, coalesced writes ----
        for (int e = t; e < RA_CCHUNK * 7 * 7; e += 256) {
            const int c  = e / 49;
            const int r  = e - c * 49;
            const int py = r / 7;
            const int px = r - py * 7;
            const float m00 = sV[c][py][px];
            const float m01 = sV[c][py][px + 1];
            const float m10 = sV[c][py + 1][px];
            const float m11 = sV[c][py + 1][px + 1];
            const float m = fmaxf(fmaxf(m00, m01), fmaxf(m10, m11));
            out[((size_t)n * RA_C + chunk + c) * 49 + r] = m;
        }
    }
}

extern "C" void kernel_launch(void* const* d_in, const int* in_sizes, int n_in,
                              void* d_out, int out_size, void* d_ws, size_t ws_size,
                              hipStream_t stream) {
    const float* feat = (const float*)d_in[0];
    const float* rois = (const float*)d_in[1];
    float* out = (float*)d_out;
    const int nrois = in_sizes[1] / 5;
    dim3 grid((unsigned)nrois), block(256);
    hipLaunchKernelGGL(roialign_maxpool_wmma, grid, block, 0, stream,
                       feat, rois, out, nrois);
}